// BiCGSTABModel_3083786518614
// MI455X (gfx1250) — compile-verified
//
#include <hip/hip_runtime.h>
#include <hip/hip_bf16.h>

// Problem constants: [B, H, W, Z] = [2, 256, 256, 256], f32, 4 BiCGSTAB iters.
#define EPSF 1e-6f
#define NTOT    33554432ull  // 2*256^3 elements per field
#define NLINES  131072       // B*H*W contiguous z-lines
#define NBSTEN  32768        // stencil blocks (4 lines per block)
#define PPB     16384        // reduction partials per batch
#define LSTRIDE 264          // LDS floats per line (256 data + halos + 16B-align pad)

// ---------------- CDNA5 async global->LDS staging -------------------------
#if defined(__AMDGCN__) && __has_builtin(__builtin_amdgcn_global_load_async_to_lds_b128)
#define ASYNC_B128 1
#else
#define ASYNC_B128 0
#endif
#if defined(__AMDGCN__) && __has_builtin(__builtin_amdgcn_global_load_async_to_lds_b32)
#define ASYNC_B32 1
#else
#define ASYNC_B32 0
#endif

#if ASYNC_B128 || ASYNC_B32
typedef __attribute__((address_space(1))) int as_glob_i32;
typedef __attribute__((address_space(3))) int as_lds_i32;
typedef int v4i __attribute__((vector_size(16)));
typedef __attribute__((address_space(1))) v4i as_glob_v4i;
typedef __attribute__((address_space(3))) v4i as_lds_v4i;
__device__ __forceinline__ void wait_async0() {
#if __has_builtin(__builtin_amdgcn_s_wait_asynccnt)
  __builtin_amdgcn_s_wait_asynccnt(0);
#else
  asm volatile("s_wait_asynccnt 0" ::: "memory");
#endif
}
#endif

// Stage 4 z-lines (4*256 floats, contiguous in global) into a padded LDS tile
// lds[4][LSTRIDE]: data at [l][4..259], zero halos at [l][3] and [l][260].
// Data starts at byte offset l*1056+16 -> every thread's 16B chunk is aligned.
__device__ __forceinline__ void stage_tile(float* lds, const float* __restrict__ g) {
  const int t = threadIdx.x;          // 256 threads
  const int l = t >> 6, q = t & 63;   // sub-line, 16B-chunk within line
  float* dst = lds + l * LSTRIDE + 4 + 4 * q;
  const float* src = g + l * 256 + 4 * q;
#if ASYNC_B128
  __builtin_amdgcn_global_load_async_to_lds_b128(
      (as_glob_v4i*)src, (as_lds_v4i*)dst, 0, 0);
#elif ASYNC_B32
#pragma unroll
  for (int k = 0; k < 4; ++k)
    __builtin_amdgcn_global_load_async_to_lds_b32(
        (as_glob_i32*)(src + k), (as_lds_i32*)(dst + k), 0, 0);
#else
  *(float4*)dst = *(const float4*)src;
#endif
  if (t < 8) {                        // zero the 8 halo slots
    const int ll = t >> 1, side = t & 1;
    lds[ll * LSTRIDE + (side ? 260 : 3)] = 0.0f;
  }
#if ASYNC_B128 || ASYNC_B32
  wait_async0();
#endif
  __syncthreads();
}

// ---------------- small float4 helpers ------------------------------------
__device__ __forceinline__ void sub4(float4& a, const float4 b) {
  a.x -= b.x; a.y -= b.y; a.z -= b.z; a.w -= b.w;
}
__device__ __forceinline__ float dot4(const float4 a, const float4 b) {
  return a.x * b.x + a.y * b.y + a.z * b.z + a.w * b.w;
}

// ---------------- deterministic block reduction (wave32) ------------------
__device__ __forceinline__ float block_reduce(float val) {
  __shared__ float sm[32];
  for (int o = 16; o > 0; o >>= 1) val += __shfl_down(val, o, 32);
  const int lane = threadIdx.x & 31;
  const int wid  = threadIdx.x >> 5;
  const int nw   = blockDim.x >> 5;
  if (lane == 0) sm[wid] = val;
  __syncthreads();
  if (wid == 0) {
    val = (lane < nw) ? sm[lane] : 0.0f;
    for (int o = 16; o > 0; o >>= 1) val += __shfl_down(val, o, 32);
  }
  __syncthreads();   // make sm reusable by a second call
  return val;        // valid in thread 0
}

// ---------------- 7-point stencil over one float4 -------------------------
// Own line from LDS (branchless z halos); w/h halo lines as float4 from L2.
__device__ __forceinline__ float4 stencil4(const float* __restrict__ f,
                                           const float* myl, float c,
                                           int q, int w, int h, size_t base,
                                           float4& xc_out) {
  const float4 xc = *(const float4*)myl;
  const float zl = myl[-1];   // halo-safe: q==0 hits the zero slot
  const float zr = myl[4];    // halo-safe: q==63 hits the zero slot
  float4 acc;
  acc.x = c * xc.x - zl   - xc.y;
  acc.y = c * xc.y - xc.x - xc.z;
  acc.z = c * xc.z - xc.y - xc.w;
  acc.w = c * xc.w - xc.z - zr;
  const size_t o = base + 4 * (size_t)q;
  if (w > 0)   sub4(acc, *(const float4*)(f + o - 256));
  if (w < 255) sub4(acc, *(const float4*)(f + o + 256));
  if (h > 0)   sub4(acc, *(const float4*)(f + o - 65536));
  if (h < 255) sub4(acc, *(const float4*)(f + o + 65536));
  xc_out = xc;
  return acc;
}

#define STEN_PROLOGUE(fld)                                   \
  __shared__ float lds[4 * LSTRIDE];                         \
  const int t = threadIdx.x, l = t >> 6, q = t & 63;         \
  const int L = blockIdx.x * 4 + l;                          \
  const int w = L & 255, h = (L >> 8) & 255;                 \
  const size_t base = (size_t)L << 8;                        \
  const size_t o = base + 4 * (size_t)q;                     \
  stage_tile(lds, (fld) + ((size_t)blockIdx.x << 10));       \
  const float c = center[L & 65535];                         \
  const float* myl = lds + l * LSTRIDE + 4 + 4 * q;          \
  float4 xc;

// r0 = b - stencil(x); r = p = r0; P0[blk] = block-partial of dot(r0, r0)
__global__ void k_init(const float* __restrict__ x, const float* __restrict__ bb,
                       const float* __restrict__ center,
                       float* __restrict__ r0, float* __restrict__ r,
                       float* __restrict__ p, float* __restrict__ P0) {
  STEN_PROLOGUE(x)
  const float4 st = stencil4(x, myl, c, q, w, h, base, xc);
  const float4 bv = *(const float4*)(bb + o);
  float4 rv;
  rv.x = bv.x - st.x; rv.y = bv.y - st.y; rv.z = bv.z - st.z; rv.w = bv.w - st.w;
  *(float4*)(r0 + o) = rv;
  *(float4*)(r  + o) = rv;
  *(float4*)(p  + o) = rv;
  const float part = block_reduce(dot4(rv, rv));
  if (t == 0) P0[blockIdx.x] = part;
}

// v = stencil(p); P0[blk] = partial of dot(r0, v)
__global__ void k_spmv(const float* __restrict__ p, const float* __restrict__ center,
                       const float* __restrict__ r0, float* __restrict__ v,
                       float* __restrict__ P0) {
  STEN_PROLOGUE(p)
  const float4 out = stencil4(p, myl, c, q, w, h, base, xc);
  *(float4*)(v + o) = out;
  const float4 r0v = *(const float4*)(r0 + o);
  const float part = block_reduce(dot4(r0v, out));
  if (t == 0) P0[blockIdx.x] = part;
}

// t = stencil(s); P0[blk] = partial dot(t,s); P1[blk] = partial dot(t,t)
__global__ void k_spmv2(const float* __restrict__ s, const float* __restrict__ center,
                        float* __restrict__ tt, float* __restrict__ P0,
                        float* __restrict__ P1) {
  STEN_PROLOGUE(s)
  const float4 out = stencil4(s, myl, c, q, w, h, base, xc);
  *(float4*)(tt + o) = out;
  const float pts = block_reduce(dot4(out, xc));   // s center value from LDS
  if (t == 0) P0[blockIdx.x] = pts;
  const float ptt = block_reduce(dot4(out, out));
  if (t == 0) P1[blockIdx.x] = ptt;
}

// s = r - alpha*v   (float4, per-batch alpha from S[2+b])
__global__ void k_saxpy_s(const float4* __restrict__ r4, const float4* __restrict__ v4,
                          const float* __restrict__ S, float4* __restrict__ s4) {
  const size_t i = (size_t)blockIdx.x * blockDim.x + threadIdx.x;
  const int b = (int)(blockIdx.x >> 14);   // 16384 blocks per batch
  const float a = S[2 + b];
  const float4 rv = r4[i], vv = v4[i];
  float4 oo;
  oo.x = rv.x - a * vv.x;  oo.y = rv.y - a * vv.y;
  oo.z = rv.z - a * vv.z;  oo.w = rv.w - a * vv.w;
  s4[i] = oo;
}

// x += alpha*p + omega*s;  r = s - omega*t;  P0[blk] = partial dot(r0, r)
__global__ void k_xr(float4* __restrict__ x4, const float4* __restrict__ p4,
                     const float4* __restrict__ s4, const float4* __restrict__ t4,
                     const float4* __restrict__ r04, const float* __restrict__ S,
                     float4* __restrict__ r4, float* __restrict__ P0) {
  const size_t i = (size_t)blockIdx.x * 256 + threadIdx.x;
  const int b = (int)(blockIdx.x >> 14);
  const float alpha = S[2 + b], omega = S[4 + b];
  const float4 sv = s4[i], tv = t4[i], pv = p4[i], r0v = r04[i];
  float4 xv = x4[i], rv;
  xv.x += alpha * pv.x + omega * sv.x;  rv.x = sv.x - omega * tv.x;
  xv.y += alpha * pv.y + omega * sv.y;  rv.y = sv.y - omega * tv.y;
  xv.z += alpha * pv.z + omega * sv.z;  rv.z = sv.z - omega * tv.z;
  xv.w += alpha * pv.w + omega * sv.w;  rv.w = sv.w - omega * tv.w;
  x4[i] = xv;
  r4[i] = rv;
  const float part = block_reduce(dot4(r0v, rv));
  if (threadIdx.x == 0) P0[blockIdx.x] = part;
}

// p = r + beta*(p - omega*v)   (float4, per-batch omega/beta)
__global__ void k_pupd(const float4* __restrict__ r4, const float4* __restrict__ v4,
                       const float* __restrict__ S, float4* __restrict__ p4) {
  const size_t i = (size_t)blockIdx.x * blockDim.x + threadIdx.x;
  const int b = (int)(blockIdx.x >> 14);
  const float omega = S[4 + b], beta = S[6 + b];
  const float4 rv = r4[i], vv = v4[i], pv = p4[i];
  float4 oo;
  oo.x = rv.x + beta * (pv.x - omega * vv.x);
  oo.y = rv.y + beta * (pv.y - omega * vv.y);
  oo.z = rv.z + beta * (pv.z - omega * vv.z);
  oo.w = rv.w + beta * (pv.w - omega * vv.w);
  p4[i] = oo;
}

// ---------------- scalar reduction kernels (2 blocks = 2 batches) ---------
// S layout: [0..1]=rho, [2..3]=alpha, [4..5]=omega, [6..7]=beta
__device__ __forceinline__ float sum_partials(const float* P, int b) {
  float acc = 0.f;
  for (int j = threadIdx.x; j < PPB; j += 1024) acc += P[b * PPB + j];
  return block_reduce(acc);
}

__global__ void r_rho(const float* __restrict__ P0, float* __restrict__ S) {
  const int b = blockIdx.x;
  const float tot = sum_partials(P0, b);
  if (threadIdx.x == 0) S[b] = tot;                       // rho = dot(r0,r0)
}
__global__ void r_alpha(const float* __restrict__ P0, float* __restrict__ S) {
  const int b = blockIdx.x;
  const float tot = sum_partials(P0, b);                  // dot(r0, v)
  if (threadIdx.x == 0) S[2 + b] = S[b] / (tot + EPSF);
}
__global__ void r_omega(const float* __restrict__ P0, const float* __restrict__ P1,
                        float* __restrict__ S) {
  const int b = blockIdx.x;
  const float ts = sum_partials(P0, b);
  const float tw = sum_partials(P1, b);
  if (threadIdx.x == 0) S[4 + b] = ts / (tw + EPSF);
}
__global__ void r_beta(const float* __restrict__ P0, float* __restrict__ S) {
  const int b = blockIdx.x;
  const float rho_next = sum_partials(P0, b);             // dot(r0, r)
  if (threadIdx.x == 0) {
    const float beta = rho_next / (S[b] + EPSF) * (S[2 + b] / (S[4 + b] + EPSF));
    S[6 + b] = beta;
    S[b] = rho_next;
  }
}

// ---------------- driver ---------------------------------------------------
extern "C" void kernel_launch(void* const* d_in, const int* in_sizes, int n_in,
                              void* d_out, int out_size, void* d_ws, size_t ws_size,
                              hipStream_t stream) {
  (void)in_sizes; (void)n_in; (void)out_size; (void)ws_size;
  const float* x0 = (const float*)d_in[0];
  const float* bb = (const float*)d_in[1];
  const float* ct = (const float*)d_in[3];   // d_in[2] ("ref") unused by reference
  float* x = (float*)d_out;

  float* r0 = (float*)d_ws;                  // 6 fields + partials + scalars
  float* r  = r0 + NTOT;
  float* p  = r0 + 2 * NTOT;
  float* v  = r0 + 3 * NTOT;
  float* s  = r0 + 4 * NTOT;
  float* t  = r0 + 5 * NTOT;
  float* P0 = r0 + 6 * NTOT;
  float* P1 = P0 + NBSTEN;
  float* S  = P1 + NBSTEN;

  (void)hipMemcpyAsync(x, x0, NTOT * sizeof(float), hipMemcpyDeviceToDevice, stream);

  const dim3 gl(NBSTEN), bl(256);
  k_init<<<gl, bl, 0, stream>>>(x0, bb, ct, r0, r, p, P0);
  r_rho<<<2, 1024, 0, stream>>>(P0, S);

  for (int it = 0; it < 4; ++it) {
    k_spmv<<<gl, bl, 0, stream>>>(p, ct, r0, v, P0);
    r_alpha<<<2, 1024, 0, stream>>>(P0, S);
    k_saxpy_s<<<NBSTEN, 256, 0, stream>>>((const float4*)r, (const float4*)v, S, (float4*)s);
    k_spmv2<<<gl, bl, 0, stream>>>(s, ct, t, P0, P1);
    r_omega<<<2, 1024, 0, stream>>>(P0, P1, S);
    k_xr<<<gl, bl, 0, stream>>>((float4*)x, (const float4*)p, (const float4*)s,
                                (const float4*)t, (const float4*)r0, S,
                                (float4*)r, P0);
    if (it < 3) {                    // output is x only; skip dead final p-update
      r_beta<<<2, 1024, 0, stream>>>(P0, S);
      k_pupd<<<NBSTEN, 256, 0, stream>>>((const float4*)r, (const float4*)v, S, (float4*)p);
    }
  }
}